// HGNN_ATT_74251394613676
// MI455X (gfx1250) — compile-verified
//
#include <hip/hip_runtime.h>
#include <cstdint>
#include <cstddef>

// ---------------- problem constants (match reference) ----------------
constexpr int NU_ = 50000;
constexpr int NN  = 100000;   // NU + NI
constexpr int DD  = 64;
constexpr int KK  = 4;
constexpr int EE  = 1000000;

typedef float v2f __attribute__((ext_vector_type(2)));
typedef float v8f __attribute__((ext_vector_type(8)));

// Relaxed, agent-scope fp32 add -> should lower to hardware
// global_atomic_add_f32 (no-return, STOREcnt-tracked) on gfx1250,
// instead of a CAS loop.
__device__ __forceinline__ void gAtomicAddF32(float* p, float v) {
    __hip_atomic_fetch_add(p, v, __ATOMIC_RELAXED, __HIP_MEMORY_SCOPE_AGENT);
}

// ---------------- small utility kernels ----------------
__global__ void hgnn_zero4_kernel(float4* __restrict__ p, int n4) {
    int i = blockIdx.x * blockDim.x + threadIdx.x;
    if (i < n4) p[i] = make_float4(0.f, 0.f, 0.f, 0.f);
}

// deg[src] += 1 for src = rows  and  src = cols + NU
__global__ void hgnn_deg_kernel(const int* __restrict__ rows, const int* __restrict__ cols,
                                float* __restrict__ deg, int E) {
    int i = blockIdx.x * blockDim.x + threadIdx.x;
    if (i < E) {
        gAtomicAddF32(&deg[rows[i]], 1.0f);
        gAtomicAddF32(&deg[cols[i] + NU_], 1.0f);
    }
}

__global__ void hgnn_rsqrt_kernel(float* __restrict__ d, int n) {
    int i = blockIdx.x * blockDim.x + threadIdx.x;
    if (i < n) d[i] = rsqrtf(d[i] + 1e-7f);
}

// h = relu(x) + bias[d]
__global__ void hgnn_relu_bias_kernel(const float* __restrict__ x, const float* __restrict__ bias,
                                      float* __restrict__ h, int total) {
    int i = blockIdx.x * blockDim.x + threadIdx.x;
    if (i < total) h[i] = fmaxf(x[i], 0.0f) + bias[i & (DD - 1)];
}

// One wave (32 lanes) per interaction; both symmetric directions.
//   dst[r]     += val * src[c+NU]
//   dst[c+NU]  += val * src[r]
// 256 B coalesced row gathers (L2-resident), fp32 hw atomics to L2.
__global__ __launch_bounds__(256)
void hgnn_spmm_kernel(const int* __restrict__ rows, const int* __restrict__ cols,
                      const float* __restrict__ dinv,
                      const float* __restrict__ src, float* __restrict__ dst, int E) {
    int wid  = (int)((blockIdx.x * (unsigned)blockDim.x + threadIdx.x) >> 5);
    int lane = threadIdx.x & 31;
    if (wid >= E) return;
    int r = rows[wid];
    int c = cols[wid] + NU_;
    float val = dinv[r] * dinv[c];
    const float2 vr = *(const float2*)(src + (size_t)r * DD + lane * 2);
    const float2 vc = *(const float2*)(src + (size_t)c * DD + lane * 2);
    float* dr = dst + (size_t)r * DD + lane * 2;
    float* dc = dst + (size_t)c * DD + lane * 2;
    gAtomicAddF32(dr,     val * vc.x);
    gAtomicAddF32(dr + 1, val * vc.y);
    gAtomicAddF32(dc,     val * vr.x);
    gAtomicAddF32(dc + 1, val * vr.y);
}

// Row softmax over D=64, one wave per row, two elements per lane.
__global__ __launch_bounds__(256)
void hgnn_softmax_kernel(float* __restrict__ node, int nRows) {
    int row  = (int)((blockIdx.x * (unsigned)blockDim.x + threadIdx.x) >> 5);
    int lane = threadIdx.x & 31;
    if (row >= nRows) return;
    float* p = node + (size_t)row * DD;
    float a = p[lane], b = p[lane + 32];
    float m = fmaxf(a, b);
    #pragma unroll
    for (int off = 16; off > 0; off >>= 1) m = fmaxf(m, __shfl_xor(m, off, 32));
    float ea = expf(a - m), eb = expf(b - m);
    float s = ea + eb;
    #pragma unroll
    for (int off = 16; off > 0; off >>= 1) s += __shfl_xor(s, off, 32);
    float inv = 1.0f / s;
    p[lane]      = ea * inv;
    p[lane + 32] = eb * inv;
}

// ---------------- fused fc1 + fusion-gate tail (WMMA f32 16x16x4) ----------------
// Block = 128 threads (4 waves), one 16-row tile of nodes.
// wave w owns output-column tile j0 = 16*w.
//
// f32 WMMA operand layouts (ISA 7.12.2):
//   A 16x4 : lanes 0-15 rows M; vgpr pair = {K=kb, K=kb+1} (lanes<16) / {kb+2, kb+3} (lanes>=16)
//   B 4x16 : mirrored — lane n holds W[j0+n][kb..kb+1] (lanes<16) / [kb+2..kb+3] (lanes>=16)
//   C/D    : vgpr r = row (r + 8*half), lane&15 = column
__global__ __launch_bounds__(128)
void hgnn_tail_kernel(const float* __restrict__ P,   // softmaxed node  (N x 64)
                      const float* __restrict__ Xc,  // current x       (N x 64)
                      const float* __restrict__ Wfc, // fc1_W   (64 x 64), out[j]=sum_k in[k]*W[j][k]
                      const float* __restrict__ W1,  // fus_l1_W(64 x 64)
                      const float* __restrict__ B1,  // fus_l1_b(64)
                      const float* __restrict__ W2,  // fus_l2_W(64)
                      float* __restrict__ out) {     // x_new (N x 64)
    __shared__ float sP[16][68];
    __shared__ float sX[16][68];
    __shared__ float sF[16][68];
    __shared__ float sS[2][16];

    const int tid  = threadIdx.x;
    const int wave = tid >> 5;
    const int lane = tid & 31;
    const int half = lane >> 4;   // 0 or 1
    const int ln   = lane & 15;
    const size_t m0 = (size_t)blockIdx.x * 16;

    // stage P and Xc tiles (8 floats per thread, float4 loads)
    {
        int r  = tid >> 3;        // 0..15
        int c0 = (tid & 7) * 8;   // 0..56
        const float4* pP = (const float4*)(P  + (m0 + r) * DD + c0);
        const float4* pX = (const float4*)(Xc + (m0 + r) * DD + c0);
        float4 a0 = pP[0], a1 = pP[1];
        float4 x0 = pX[0], x1 = pX[1];
        sP[r][c0+0]=a0.x; sP[r][c0+1]=a0.y; sP[r][c0+2]=a0.z; sP[r][c0+3]=a0.w;
        sP[r][c0+4]=a1.x; sP[r][c0+5]=a1.y; sP[r][c0+6]=a1.z; sP[r][c0+7]=a1.w;
        sX[r][c0+0]=x0.x; sX[r][c0+1]=x0.y; sX[r][c0+2]=x0.z; sX[r][c0+3]=x0.w;
        sX[r][c0+4]=x1.x; sX[r][c0+5]=x1.y; sX[r][c0+6]=x1.z; sX[r][c0+7]=x1.w;
    }
    if (tid < 32) sS[tid >> 4][tid & 15] = 0.0f;
    __syncthreads();

    const int j0 = wave * 16;

    // ---- Phase 1: F = P @ Wfc^T, this wave computes F[:, j0:j0+16] ----
    v8f acc = {0.f,0.f,0.f,0.f,0.f,0.f,0.f,0.f};
    #pragma unroll
    for (int c = 0; c < 16; ++c) {
        const int kb = 4 * c + 2 * half;
        const float2 av = *(const float2*)&sP[ln][kb];
        const float2 bw = *(const float2*)(Wfc + (j0 + ln) * DD + kb);
        v2f a; a.x = av.x; a.y = av.y;
        v2f b; b.x = bw.x; b.y = bw.y;
        acc = __builtin_amdgcn_wmma_f32_16x16x4_f32(false, a, false, b, (short)0, acc, false, false);
    }
    #pragma unroll
    for (int r = 0; r < 8; ++r) sF[r + 8 * half][j0 + ln] = acc[r];
    __syncthreads();

    // ---- Phase 2: branch scores s_b[m] = sum_j W2[j] * tanh((emb_b @ W1^T)[m,j] + B1[j]) ----
    const float bj = B1[j0 + ln];
    const float wj = W2[j0 + ln];
    #pragma unroll
    for (int br = 0; br < 2; ++br) {
        const float (*S)[68] = (br == 0) ? sX : sF;   // branch 0 = xc, branch 1 = fc1(node)
        v8f g = {0.f,0.f,0.f,0.f,0.f,0.f,0.f,0.f};
        #pragma unroll
        for (int c = 0; c < 16; ++c) {
            const int kb = 4 * c + 2 * half;
            const float2 av = *(const float2*)&S[ln][kb];
            const float2 bw = *(const float2*)(W1 + (j0 + ln) * DD + kb);
            v2f a; a.x = av.x; a.y = av.y;
            v2f b; b.x = bw.x; b.y = bw.y;
            g = __builtin_amdgcn_wmma_f32_16x16x4_f32(false, a, false, b, (short)0, g, false, false);
        }
        float part[8];
        #pragma unroll
        for (int r = 0; r < 8; ++r) part[r] = wj * tanhf(g[r] + bj);
        // reduce over the 16 lanes of this half (xor 1,2,4,8 stays within half)
        #pragma unroll
        for (int off = 1; off < 16; off <<= 1) {
            #pragma unroll
            for (int r = 0; r < 8; ++r) part[r] += __shfl_xor(part[r], off, 32);
        }
        if (ln == 0) {
            #pragma unroll
            for (int r = 0; r < 8; ++r) atomicAdd(&sS[br][r + 8 * half], part[r]);
        }
    }
    __syncthreads();

    // ---- Phase 3: 2-way softmax gate (fus_l2_b cancels) + blend + writeout ----
    {
        int r  = tid >> 3;
        int c0 = (tid & 7) * 8;
        float s0 = sS[0][r], s1 = sS[1][r];
        float mx = fmaxf(s0, s1);
        float e0 = expf(s0 - mx), e1 = expf(s1 - mx);
        float inv = 1.0f / (e0 + e1);
        float sc0 = e0 * inv, sc1 = e1 * inv;
        float* o = out + (m0 + r) * DD + c0;
        #pragma unroll
        for (int c = 0; c < 8; ++c) o[c] = sc0 * sX[r][c0 + c] + sc1 * sF[r][c0 + c];
    }
}

// ---------------- host-side orchestration ----------------
extern "C" void kernel_launch(void* const* d_in, const int* in_sizes, int n_in,
                              void* d_out, int out_size, void* d_ws, size_t ws_size,
                              hipStream_t stream) {
    const float* x         = (const float*)d_in[0];
    const float* hgc1_bias = (const float*)d_in[1];
    const float* fc1_W     = (const float*)d_in[2];
    const float* fus_l1_W  = (const float*)d_in[3];
    const float* fus_l1_b  = (const float*)d_in[4];
    const float* fus_l2_W  = (const float*)d_in[5];
    // d_in[6] = fus_l2_b: added to both branches -> cancels in the 2-way softmax
    const int* rows = (const int*)d_in[7];
    const int* cols = (const int*)d_in[8];

    const int ND = NN * DD;
    float* nodesOut = (float*)d_out;                      // (K, N, D)
    float* edgesOut = (float*)d_out + (size_t)KK * ND;    // (K, N, D)

    // workspace: h/node buffer (N*D) reused across the two SpMMs, + dinv (N)
    float* hbuf = (float*)d_ws;
    float* dinv = hbuf + (size_t)ND;

    const dim3 b256(256);
    const int gND   = (ND / 4 + 255) / 256;   // float4 zeroing of N*D
    const int gN4   = (NN / 4 + 255) / 256;   // float4 zeroing of N
    const int gN    = (NN + 255) / 256;
    const int gNDe  = (ND + 255) / 256;
    const int gE    = (EE + 255) / 256;
    const int gWav  = (EE + 7) / 8;           // 8 waves / 256-thread block, wave per nnz pair
    const int gRow  = (NN + 7) / 8;           // 8 rows / block for softmax
    const int gTile = NN / 16;                // 16-row tiles for the WMMA tail

    for (int k = 0; k < KK; ++k) {
        const float* xc = (k == 0) ? x : (nodesOut + (size_t)(k - 1) * ND);
        const int* rk = rows + (size_t)k * EE;
        const int* ck = cols + (size_t)k * EE;
        float* edgek  = edgesOut + (size_t)k * ND;
        float* nodek  = nodesOut + (size_t)k * ND;

        // degree -> dinv
        hgnn_zero4_kernel<<<gN4, b256, 0, stream>>>((float4*)dinv, NN / 4);
        hgnn_deg_kernel<<<gE, b256, 0, stream>>>(rk, ck, dinv, EE);
        hgnn_rsqrt_kernel<<<gN, b256, 0, stream>>>(dinv, NN);

        // h = relu(xc) + bias
        hgnn_relu_bias_kernel<<<gNDe, b256, 0, stream>>>(xc, hgc1_bias, hbuf, ND);

        // edge = L @ h   (accumulate directly into edges[k] output region)
        hgnn_zero4_kernel<<<gND, b256, 0, stream>>>((float4*)edgek, ND / 4);
        hgnn_spmm_kernel<<<gWav, b256, 0, stream>>>(rk, ck, dinv, hbuf, edgek, EE);

        // node = L @ edge  (reuse hbuf as node buffer)
        hgnn_zero4_kernel<<<gND, b256, 0, stream>>>((float4*)hbuf, ND / 4);
        hgnn_spmm_kernel<<<gWav, b256, 0, stream>>>(rk, ck, dinv, edgek, hbuf, EE);

        // row softmax, then fc1 + fusion gate -> x_new = nodes[k]
        hgnn_softmax_kernel<<<gRow, b256, 0, stream>>>(hbuf, NN);
        hgnn_tail_kernel<<<gTile, dim3(128), 0, stream>>>(hbuf, xc, fc1_W, fus_l1_W,
                                                          fus_l1_b, fus_l2_W, nodek);
    }
}